// BrushStrokeRenderer_78176994722066
// MI455X (gfx1250) — compile-verified
//
#include <hip/hip_runtime.h>
#include <math.h>

#define NSTROKE 16
#define NCP     7
#define NG      4
#define WC      320
#define HC      320
#define FEATHERF 2.0f
#define EPSF     0.01f

typedef float v2f __attribute__((ext_vector_type(2)));
typedef float v8f __attribute__((ext_vector_type(8)));

__device__ __forceinline__ float clamp01f(float x) { return fminf(fmaxf(x, 0.f), 1.f); }
__device__ __forceinline__ float horner4(float c0, float c1, float c2, float c3, float t) {
  return fmaf(fmaf(fmaf(c3, t, c2), t, c1), t, c0);
}

// ---------------------------------------------------------------------------
// Kernel 1: coefficient einsum  coe[c,s,g,e] = sum_f MAT[e,f] * vin[c,s,g+f]
// done with V_WMMA_F32_16X16X4_F32:  D(16x16) = A(16x4) x B(4x16)
//   A rows  = 16 consecutive (c,s,g) windows, A[m][f] = vin[c,s,g+f]
//   B[f][e] = MAT[e][f] (only e<4 columns valid, rest zero)
// 256 rows total -> 16 WMMA issues in a single wave32.
// ---------------------------------------------------------------------------
__global__ __launch_bounds__(32) void coef_kernel(
    const float* __restrict__ cp, const float* __restrict__ dep,
    const float* __restrict__ wid, float* __restrict__ coe)
{
  const int lane = threadIdx.x;
  const int half = lane >> 4;   // selects K-pair (A) / K rows (B) and M-half (D)
  const int m    = lane & 15;
  const int k0   = half * 2;

  // MAT (already scaled by 0.5), row e, col f
  const float matv[4][4] = {
    { 0.0f,  1.0f,  0.0f,  0.0f},
    {-0.5f,  0.0f,  0.5f,  0.0f},
    { 1.0f, -2.5f,  2.0f, -0.5f},
    {-0.5f,  1.5f, -1.5f,  0.5f}};

  // B layout (4x16 f32): VGPR r, lane-half h -> K = 2h + r, N = lane%16
  v2f b;
  b.x = (m < 4) ? matv[m][k0 + 0] : 0.f;   // B[K][N] = MAT[N][K]
  b.y = (m < 4) ? matv[m][k0 + 1] : 0.f;

  for (int wblk = 0; wblk < 16; ++wblk) {
    const int R = wblk * 16 + m;       // row id = c*64 + s*4 + g
    const int c = R >> 6;
    const int s = (R >> 2) & 15;
    const int g = R & 3;
    float a0, a1;
    if (c < 2) {
      a0 = cp[(c * NSTROKE + s) * NCP + g + k0];
      a1 = cp[(c * NSTROKE + s) * NCP + g + k0 + 1];
    } else if (c == 2) {
      a0 = dep[s * NCP + g + k0];
      a1 = dep[s * NCP + g + k0 + 1];
    } else {
      a0 = wid[s * NCP + g + k0];
      a1 = wid[s * NCP + g + k0 + 1];
    }
    v2f a; a.x = a0; a.y = a1;         // A layout: VGPR r, half h -> K = 2h + r, M = lane%16
    v8f acc = {0.f, 0.f, 0.f, 0.f, 0.f, 0.f, 0.f, 0.f};
    acc = __builtin_amdgcn_wmma_f32_16x16x4_f32(false, a, false, b,
                                                (short)0, acc, false, false);
    // D layout: VGPR i, half h -> M = i + 8h, N = lane%16; keep N<4 (valid e)
    if (m < 4) {
      #pragma unroll
      for (int i = 0; i < 8; ++i)
        coe[(wblk * 16 + half * 8 + i) * 4 + m] = acc[i];
    }
  }
}

// ---------------------------------------------------------------------------
// Kernel 2: per-pixel Newton root refinement + argmin + depth sort + composite
// ---------------------------------------------------------------------------
__global__ __launch_bounds__(64) void render_kernel(
    const float* __restrict__ coe, const float* __restrict__ color,
    const float* __restrict__ noise, float* __restrict__ out)
{
  __shared__ float sCoe[1024];            // [c][s][g][e] : c*256 + s*16 + g*4 + e
  __shared__ float sCol[64];              // [s][4]
  __shared__ float sKey[NSTROKE][64];     // per-thread column: depth per stroke
  __shared__ float sAlp[NSTROKE][64];     // per-thread column: alpha per stroke

  const int tid = threadIdx.x;
  #pragma unroll
  for (int i = 0; i < 16; ++i) sCoe[i * 64 + tid] = coe[i * 64 + tid];
  sCol[tid] = color[tid];
  __syncthreads();

  const int u = blockIdx.x * 64 + tid;    // axis 0 of (U,V); contiguous in output
  const int v = blockIdx.y;
  const float uf = (float)u, vf = (float)v;
  const float nz = noise[u * WC + v];

  #pragma unroll 1
  for (int s = 0; s < NSTROKE; ++s) {
    float bestd = 1e30f, bestt = 0.f;
    int   bestg = 0;
    #pragma unroll 1
    for (int g = 0; g < NG; ++g) {
      const int base = s * 16 + g * 4;    // uniform across lanes -> LDS broadcast
      const float cx0 = sCoe[base + 0], cx1 = sCoe[base + 1],
                  cx2 = sCoe[base + 2], cx3 = sCoe[base + 3];
      const float cy0 = sCoe[256 + base + 0], cy1 = sCoe[256 + base + 1],
                  cy2 = sCoe[256 + base + 2], cy3 = sCoe[256 + base + 3];
      const float cw0 = sCoe[768 + base + 0], cw1 = sCoe[768 + base + 1],
                  cw2 = sCoe[768 + base + 2], cw3 = sCoe[768 + base + 3];
      auto distF = [&](float t) -> float {
        float X = horner4(cx0, cx1, cx2, cx3, t);
        float Y = horner4(cy0, cy1, cy2, cy3, t);
        float W = fminf(fmaxf(horner4(cw0, cw1, cw2, cw3, t), 0.f), 128.f);
        float DX = X - vf, DY = Y - uf;
        return fmaf(DX, DX, fmaf(DY, DY, -W * W));
      };
      #pragma unroll
      for (int r = 0; r < 5; ++r) {
        float t = 0.1f + 0.2f * (float)r;
        #pragma unroll
        for (int it = 0; it < 3; ++it) {
          float d0 = distF(t);
          float dp = distF(t + EPSF);
          float dm = distF(t - EPSF);
          float delta = (dp - dm) / (dp + dm - 2.f * d0) * (EPSF * 0.5f);
          t = clamp01f(t - delta);
        }
        float df = distF(t);
        if (df < bestd) { bestd = df; bestt = t; bestg = g; }
      }
    }
    // final evaluation at the winning (segment, root)
    const int base = s * 16 + bestg * 4;  // per-lane index -> indexed LDS read
    const float t = bestt;
    float x = horner4(sCoe[base + 0], sCoe[base + 1], sCoe[base + 2], sCoe[base + 3], t);
    float y = horner4(sCoe[256 + base + 0], sCoe[256 + base + 1],
                      sCoe[256 + base + 2], sCoe[256 + base + 3], t);
    float z = horner4(sCoe[512 + base + 0], sCoe[512 + base + 1],
                      sCoe[512 + base + 2], sCoe[512 + base + 3], t);
    float w = fminf(fmaxf(horner4(sCoe[768 + base + 0], sCoe[768 + base + 1],
                                  sCoe[768 + base + 2], sCoe[768 + base + 3], t),
                          0.f), 128.f);
    float dx = x - vf, dy = y - uf;
    float dist = sqrtf(fmaf(dx, dx, dy * dy)) - w + nz;
    // smoothstep(-FEATHER, w/2, -dist)
    float tt = clamp01f((FEATHERF - dist) / (w * 0.5f + FEATHERF));
    float ss = tt * tt * (3.f - 2.f * tt);
    sAlp[s][tid] = ss * sCol[s * 4 + 3];
    sKey[s][tid] = z + (w - dist);
  }

  // pull per-stroke results back with static indices -> registers
  float kk[16], aa[16];
  int   id[16];
  #pragma unroll
  for (int i = 0; i < 16; ++i) { kk[i] = sKey[i][tid]; aa[i] = sAlp[i][tid]; id[i] = i; }

  // bitonic sort, ascending by depth; fully unrolled -> register-resident selects
  #pragma unroll
  for (int k = 2; k <= 16; k <<= 1) {
    #pragma unroll
    for (int j = k >> 1; j > 0; j >>= 1) {
      #pragma unroll
      for (int i = 0; i < 16; ++i) {
        const int l = i ^ j;
        if (l > i) {
          const bool up = ((i & k) == 0);
          const bool sw = up ? (kk[i] > kk[l]) : (kk[i] < kk[l]);
          float t0 = sw ? kk[l] : kk[i]; float t1 = sw ? kk[i] : kk[l];
          kk[i] = t0; kk[l] = t1;
          float a0 = sw ? aa[l] : aa[i]; float a1 = sw ? aa[i] : aa[l];
          aa[i] = a0; aa[l] = a1;
          int   i0 = sw ? id[l] : id[i]; int   i1 = sw ? id[i] : id[l];
          id[i] = i0; id[l] = i1;
        }
      }
    }
  }

  // front-to-back composite over white background
  float trans = 1.f, rr = 0.f, gg = 0.f, bb = 0.f;
  #pragma unroll
  for (int i = 0; i < 16; ++i) {
    const int sdx = id[i];
    const float wgt = aa[i] * trans;
    rr = fmaf(sCol[sdx * 4 + 0], wgt, rr);
    gg = fmaf(sCol[sdx * 4 + 1], wgt, gg);
    bb = fmaf(sCol[sdx * 4 + 2], wgt, bb);
    trans *= (1.f - aa[i]);
  }
  rr += trans; gg += trans; bb += trans;

  // out shape (3, V, U): out[c][v][u]
  out[(0 * HC + v) * WC + u] = rr;
  out[(1 * HC + v) * WC + u] = gg;
  out[(2 * HC + v) * WC + u] = bb;
}

extern "C" void kernel_launch(void* const* d_in, const int* in_sizes, int n_in,
                              void* d_out, int out_size, void* d_ws, size_t ws_size,
                              hipStream_t stream) {
  (void)in_sizes; (void)n_in; (void)out_size; (void)ws_size;
  const float* cp  = (const float*)d_in[0];  // control_points (1,2,16,7)
  const float* dep = (const float*)d_in[1];  // depths         (1,1,16,7)
  const float* wid = (const float*)d_in[2];  // widths         (1,1,16,7)
  const float* col = (const float*)d_in[3];  // color          (16,4)
  const float* noi = (const float*)d_in[4];  // noise          (320,320)
  float* coe = (float*)d_ws;                 // 1024 floats of spline coefficients
  float* out = (float*)d_out;                // (1,3,320,320)

  hipLaunchKernelGGL(coef_kernel, dim3(1), dim3(32), 0, stream, cp, dep, wid, coe);
  hipLaunchKernelGGL(render_kernel, dim3(WC / 64, HC), dim3(64), 0, stream,
                     coe, col, noi, out);
}